// StackedGCN_87351044866337
// MI455X (gfx1250) — compile-verified
//
#include <hip/hip_runtime.h>
#include <hip/hip_bf16.h>

typedef __attribute__((ext_vector_type(2))) float v2f;
typedef __attribute__((ext_vector_type(8))) float v8f;

#define HID 256
#define NSUB 4   // N sub-tiles (16 cols each) per wave -> 16x64 output tile

// ---------------- utility: zero fill ----------------
__global__ void gcn_zero_f32(float* __restrict__ p, size_t n) {
    size_t i = (size_t)blockIdx.x * blockDim.x + threadIdx.x;
    size_t stride = (size_t)gridDim.x * blockDim.x;
    for (; i < n; i += stride) p[i] = 0.0f;
}

// ---------------- degree count ----------------
__global__ void gcn_degree(const int* __restrict__ src, const int* __restrict__ dst,
                           float* __restrict__ deg_out, float* __restrict__ deg_in, int E) {
    int e = blockIdx.x * blockDim.x + threadIdx.x;
    if (e < E) {
        atomicAdd(&deg_out[src[e]], 1.0f);
        atomicAdd(&deg_in[dst[e]], 1.0f);
    }
}

// in-place deg -> rsqrt(max(deg,1))
__global__ void gcn_degnorm(float* __restrict__ p, int n) {
    int i = blockIdx.x * blockDim.x + threadIdx.x;
    if (i < n) p[i] = rsqrtf(fmaxf(p[i], 1.0f));
}

// ---------------- SpMM scatter: agg[dst] += h[src] * out_norm[src] ----------------
// 64 threads per edge, 4 contiguous floats (float4) per thread.
__global__ __launch_bounds__(256) void gcn_spmm(
    const float* __restrict__ h, const int* __restrict__ src, const int* __restrict__ dst,
    const float* __restrict__ out_norm, float* __restrict__ agg, int E) {
    long long t = (long long)blockIdx.x * blockDim.x + threadIdx.x;
    int e = (int)(t >> 6);
    int c = (int)(t & 63) << 2;
    if (e >= E) return;
    int s = src[e];
    int d = dst[e];
    float n = out_norm[s];
    float4 v = *(const float4*)(h + (size_t)s * HID + c);
    float* ap = agg + (size_t)d * HID + c;
    atomicAdd(ap + 0, v.x * n);
    atomicAdd(ap + 1, v.y * n);
    atomicAdd(ap + 2, v.z * n);
    atomicAdd(ap + 3, v.w * n);
}

// ---------------- WMMA GEMM: out = act((agg * in_norm[row]) @ W + bias) ----------------
// One wave32 computes a 16x64 output tile: one shared A fragment per K-step feeds
// NSUB=4 V_WMMA_F32_16X16X4_F32 issues (4x A-reuse, agg read once per 64 output
// cols instead of once per 16). Requires M % 16 == 0 (50000 = 3125*16), N = K = 256.
//
// fp32 A 16x4 layout: lanes 0-15 -> M=lane, VGPR{0,1}=K{0,1};
//                     lanes16-31 -> M=lane-16, VGPR{0,1}=K{2,3}.
// fp32 B 4x16 layout (row-striped like C/D): VGPR0 = rows {0 | 2}, VGPR1 = rows {1 | 3}
//                     split by lane half.
// fp32 C/D 16x16: VGPR r -> row (r | r+8) by lane half, col = lane&15.
__global__ __launch_bounds__(256) void gcn_gemm_wmma(
    const float* __restrict__ agg, const float* __restrict__ in_norm,
    const float* __restrict__ W, const float* __restrict__ bias,
    float* __restrict__ out, int M, int relu) {
    int gtid = blockIdx.x * blockDim.x + threadIdx.x;
    int wave = gtid >> 5;
    int lane = threadIdx.x & 31;

    const int tilesN = HID / (16 * NSUB);        // 4 wide-tiles across N
    int tM = (wave / tilesN) * 16;
    int tN = (wave % tilesN) * (16 * NSUB);
    if (tM >= M) return;

    int half = lane >> 4;                        // 0: lanes 0-15, 1: lanes 16-31
    int l15  = lane & 15;

    int arow = tM + l15;
    float an = in_norm[arow];                    // fold dst-side norm into A load
    const float* aptr = agg + (size_t)arow * HID + 2 * half;
    const float* bcol = W + tN + l15;            // row k of col sub-tile j: bcol[k*HID + 16*j]

    v8f c[NSUB];
    #pragma unroll
    for (int j = 0; j < NSUB; ++j) c[j] = (v8f){};

    #pragma unroll 4
    for (int k = 0; k < HID; k += 4) {
        v2f a = *(const v2f*)(aptr + k);         // K = k+2*half, k+1+2*half
        a = a * an;
        const float* brow0 = bcol + (size_t)(k + 2 * half) * HID;
        const float* brow1 = brow0 + HID;
        #pragma unroll
        for (int j = 0; j < NSUB; ++j) {
            v2f b;
            b.x = brow0[16 * j];
            b.y = brow1[16 * j];
            c[j] = __builtin_amdgcn_wmma_f32_16x16x4_f32(
                /*neg_a=*/false, a, /*neg_b=*/false, b,
                /*c_mod=*/(short)0, c[j], /*reuse_a=*/false, /*reuse_b=*/false);
        }
    }

    float* obase = out + (size_t)(tM + 8 * half) * HID;
    #pragma unroll
    for (int j = 0; j < NSUB; ++j) {
        int col = tN + 16 * j + l15;
        float bb = bias[col];
        float* orow = obase + col;
        #pragma unroll
        for (int r = 0; r < 8; ++r) {
            float v = c[j][r] + bb;
            if (relu) v = v > 0.0f ? v : 0.0f;
            orow[(size_t)r * HID] = v;
        }
    }
}

// ---------------- host launch ----------------
extern "C" void kernel_launch(void* const* d_in, const int* in_sizes, int n_in,
                              void* d_out, int out_size, void* d_ws, size_t ws_size,
                              hipStream_t stream) {
    const float* x  = (const float*)d_in[0];     // [N, 256]
    const float* Wa = (const float*)d_in[1];     // [3, 256, 256]
    const float* ba = (const float*)d_in[2];     // [3, 256]
    const int* src  = (const int*)d_in[3];       // [E]
    const int* dst  = (const int*)d_in[4];       // [E]

    const int N = in_sizes[0] / HID;             // 50000
    const int E = in_sizes[3];                   // 800000
    const int KHOP = in_sizes[1] / (HID * HID);  // 3

    // workspace layout (float units, 256B-aligned chunks)
    auto align_up = [](size_t v) { return (v + 63) & ~(size_t)63; };
    float* ws = (float*)d_ws;
    size_t off = 0;
    float* out_norm = ws + off; off += align_up((size_t)N);
    float* in_norm  = ws + off; off += align_up((size_t)N);
    float* agg      = ws + off; off += align_up((size_t)N * HID);
    float* hbuf     = ws + off; off += align_up((size_t)N * HID);
    (void)ws_size;

    // 1) degrees -> norms
    {
        int zb = (int)(((size_t)N + 255) / 256); if (zb > 4096) zb = 4096;
        gcn_zero_f32<<<zb, 256, 0, stream>>>(out_norm, (size_t)N);
        gcn_zero_f32<<<zb, 256, 0, stream>>>(in_norm, (size_t)N);
        gcn_degree<<<(E + 255) / 256, 256, 0, stream>>>(src, dst, out_norm, in_norm, E);
        gcn_degnorm<<<(N + 255) / 256, 256, 0, stream>>>(out_norm, N);
        gcn_degnorm<<<(N + 255) / 256, 256, 0, stream>>>(in_norm, N);
    }

    // 2) K_HOP layers
    const size_t feat_elems = (size_t)N * HID;
    const long long spmm_threads = (long long)E * 64;
    const int spmm_blocks = (int)((spmm_threads + 255) / 256);
    const int tiles = ((N + 15) / 16) * (HID / (16 * NSUB));
    const int gemm_blocks = (tiles * 32 + 255) / 256;

    const float* h_in = x;
    for (int i = 0; i < KHOP; ++i) {
        int zb = (int)((feat_elems / 4 + 255) / 256); if (zb > 8192) zb = 8192;
        gcn_zero_f32<<<zb, 256, 0, stream>>>(agg, feat_elems);
        gcn_spmm<<<spmm_blocks, 256, 0, stream>>>(h_in, src, dst, out_norm, agg, E);
        float* h_out = (i == KHOP - 1) ? (float*)d_out : hbuf;
        gcn_gemm_wmma<<<gemm_blocks, 256, 0, stream>>>(
            agg, in_norm, Wa + (size_t)i * HID * HID, ba + (size_t)i * HID,
            h_out, N, (i < KHOP - 1) ? 1 : 0);
        h_in = hbuf;
    }
}